// GraphClassification_77807627534823
// MI455X (gfx1250) — compile-verified
//
#include <hip/hip_runtime.h>
#include <hip/hip_bf16.h>

// ---------------------------------------------------------------------------
// GCN graph classification for MI455X (gfx1250, wave32).
// Bandwidth-bound workload: keep fp32 everywhere, use V_WMMA_F32_16X16X4_F32
// for the dense GEMMs, hardware global_atomic_add_f32 for the segment-sum.
// ---------------------------------------------------------------------------

typedef __attribute__((ext_vector_type(2))) float v2f;
typedef __attribute__((ext_vector_type(8))) float v8f;

#define EMBED_D   64
#define N_CENT    100
#define N_CLASS   10
#define EPS_DIST  1e-12f

__device__ __forceinline__ void atomic_add_global_f32(float* p, float v) {
  __hip_atomic_fetch_add(p, v, __ATOMIC_RELAXED, __HIP_MEMORY_SCOPE_AGENT);
}
__device__ __forceinline__ void atomic_add_lds_f32(float* p, float v) {
  __hip_atomic_fetch_add(p, v, __ATOMIC_RELAXED, __HIP_MEMORY_SCOPE_WORKGROUP);
}

// ---------------------------------------------------------------------------
// deg/dinv preparation: deg starts at 1.0 (self loop), += 1 per incoming edge,
// then dinv = rsqrt(max(deg,1)).
// ---------------------------------------------------------------------------
__global__ void init_deg_kernel(float* __restrict__ deg, int n) {
  int i = blockIdx.x * blockDim.x + threadIdx.x;
  if (i < n) deg[i] = 1.0f;
}

__global__ void accum_deg_kernel(const long long* __restrict__ dst,
                                 float* __restrict__ deg, long long e) {
  long long i = (long long)blockIdx.x * blockDim.x + threadIdx.x;
  if (i < e) atomic_add_global_f32(&deg[dst[i]], 1.0f);
}

__global__ void finish_dinv_kernel(float* __restrict__ deg, int n) {
  int i = blockIdx.x * blockDim.x + threadIdx.x;
  if (i < n) deg[i] = rsqrtf(fmaxf(deg[i], 1.0f));
}

// ---------------------------------------------------------------------------
// C[M,64] = A[M,K] @ B[64,K]^T  via V_WMMA_F32_16X16X4_F32.
// One wave -> one 16-row strip, 4 accumulators (all 64 output cols), so A is
// streamed from HBM exactly once; B (<=32KB) stays cache-resident.
// fp32 A frag 16x4: lane = m + 16*(k>=2), vgpr = k&1  -> contiguous float2.
// fp32 B frag 4x16: lane = n + 16*(k>=2), vgpr = k&1  -> row n of B, float2.
// C/D frag:        lane = n + 16*(m>=8), vgpr = m&7.
// ---------------------------------------------------------------------------
template <int K>
__global__ __launch_bounds__(256) void gemm_nt64_wmma_kernel(
    const float* __restrict__ A, const float* __restrict__ B,
    float* __restrict__ C, int M) {
  const int lane  = threadIdx.x & 31;
  const int tileM = blockIdx.x * 8 + (threadIdx.x >> 5);
  if (tileM * 16 >= M) return;              // wave-uniform exit: EXEC stays all-1s

  const int r  = lane & 15;                 // A row within tile / B row (= out col)
  const int kg = lane >> 4;                 // k-group (0 or 1)

  const float* Ap = A + (size_t)(tileM * 16 + r) * K + 2 * kg;
  const float* Bp = B + (size_t)r * K + 2 * kg;

  v8f c0 = {}, c1 = {}, c2 = {}, c3 = {};
#pragma unroll
  for (int kb = 0; kb < K; kb += 4) {
    v2f a  = *(const v2f*)(Ap + kb);
    v2f b0 = *(const v2f*)(Bp + kb);
    v2f b1 = *(const v2f*)(Bp + 16 * K + kb);
    v2f b2 = *(const v2f*)(Bp + 32 * K + kb);
    v2f b3 = *(const v2f*)(Bp + 48 * K + kb);
    c0 = __builtin_amdgcn_wmma_f32_16x16x4_f32(false, a, false, b0, (short)0, c0, false, false);
    c1 = __builtin_amdgcn_wmma_f32_16x16x4_f32(false, a, false, b1, (short)0, c1, false, false);
    c2 = __builtin_amdgcn_wmma_f32_16x16x4_f32(false, a, false, b2, (short)0, c2, false, false);
    c3 = __builtin_amdgcn_wmma_f32_16x16x4_f32(false, a, false, b3, (short)0, c3, false, false);
  }

  float* Cp = C + (size_t)(tileM * 16 + 8 * kg) * EMBED_D + r;
#pragma unroll
  for (int v = 0; v < 8; ++v) {
    Cp[(size_t)v * EMBED_D +  0] = c0[v];
    Cp[(size_t)v * EMBED_D + 16] = c1[v];
    Cp[(size_t)v * EMBED_D + 32] = c2[v];
    Cp[(size_t)v * EMBED_D + 48] = c3[v];
  }
}

// ---------------------------------------------------------------------------
// Fused gather * norm -> scatter-add (segment_sum).  64 threads per edge
// (one per feature); self loops (e >= E) folded in: src=dst=e-E, w=dinv^2.
// norm never materialized: dinv[src]*dinv[dst] recomputed (L2-resident).
// ---------------------------------------------------------------------------
__global__ void scatter_edges_kernel(const long long* __restrict__ src,
                                     const long long* __restrict__ dst,
                                     const float* __restrict__ dinv,
                                     const float* __restrict__ hw,
                                     float* __restrict__ acc,
                                     long long E, long long N) {
  long long idx = (long long)blockIdx.x * blockDim.x + threadIdx.x;
  long long total = (E + N) << 6;           // (E+N) * 64 features
  if (idx >= total) return;
  long long e = idx >> 6;
  int f = (int)(idx & 63);
  long long s, d; float w;
  if (e < E) {
    s = src[e]; d = dst[e];
    w = dinv[s] * dinv[d];
  } else {
    s = d = e - E;                          // self loop
    float di = dinv[s];
    w = di * di;
  }
  atomic_add_global_f32(&acc[(size_t)d * EMBED_D + f],
                        hw[(size_t)s * EMBED_D + f] * w);
}

// h = relu(acc + bias)
__global__ void bias_relu_kernel(const float* __restrict__ acc,
                                 const float* __restrict__ bias,
                                 float* __restrict__ h, long long total) {
  long long idx = (long long)blockIdx.x * blockDim.x + threadIdx.x;
  if (idx >= total) return;
  h[idx] = fmaxf(acc[idx] + bias[idx & (EMBED_D - 1)], 0.0f);
}

// ---------------------------------------------------------------------------
// Centroid distances + mean pooling.  Centroids staged in LDS; one thread per
// node holds its 64-f32 row in registers; per-centroid wave shuffle-reduce,
// LDS partial, one global f32 atomic per centroid per block.
// ---------------------------------------------------------------------------
__global__ __launch_bounds__(256) void centroid_pool_kernel(
    const float* __restrict__ h, const float* __restrict__ cent,
    float* __restrict__ pooled, int N) {
  __shared__ __align__(16) float s_c[N_CENT * EMBED_D];
  __shared__ float s_c2[N_CENT];
  __shared__ float s_pool[N_CENT];

  for (int i = threadIdx.x; i < N_CENT * EMBED_D; i += blockDim.x) s_c[i] = cent[i];
  for (int i = threadIdx.x; i < N_CENT; i += blockDim.x) s_pool[i] = 0.0f;
  __syncthreads();
  if (threadIdx.x < N_CENT) {
    float c2 = 0.0f;
#pragma unroll
    for (int j = 0; j < EMBED_D; ++j) {
      float v = s_c[threadIdx.x * EMBED_D + j];
      c2 += v * v;
    }
    s_c2[threadIdx.x] = c2;
  }
  __syncthreads();

  const int node = blockIdx.x * blockDim.x + threadIdx.x;
  const bool valid = node < N;
  // All lanes stay active (lane reduction below); invalid lanes read node 0.
  const float4* hp = (const float4*)(h + (size_t)(valid ? node : 0) * EMBED_D);
  float4 hr[16];
#pragma unroll
  for (int j = 0; j < 16; ++j) hr[j] = hp[j];
  float h2 = 0.0f;
#pragma unroll
  for (int j = 0; j < 16; ++j)
    h2 += hr[j].x * hr[j].x + hr[j].y * hr[j].y + hr[j].z * hr[j].z + hr[j].w * hr[j].w;

  for (int k = 0; k < N_CENT; ++k) {
    const float4* cp = (const float4*)(s_c + k * EMBED_D);
    float dot = 0.0f;
#pragma unroll
    for (int j = 0; j < 16; ++j) {
      float4 cv = cp[j];
      dot += hr[j].x * cv.x + hr[j].y * cv.y + hr[j].z * cv.z + hr[j].w * cv.w;
    }
    float d2 = fmaxf(h2 + s_c2[k] - 2.0f * dot, 0.0f);
    float dist = valid ? sqrtf(d2 + EPS_DIST) : 0.0f;
#pragma unroll
    for (int off = 16; off > 0; off >>= 1) dist += __shfl_down(dist, off, 32);
    if ((threadIdx.x & 31) == 0) atomic_add_lds_f32(&s_pool[k], dist);
  }
  __syncthreads();
  for (int i = threadIdx.x; i < N_CENT; i += blockDim.x)
    atomic_add_global_f32(&pooled[i], s_pool[i]);
}

// out[c] = b_out[c] + sum_k (pooled[k]/N) * W_out[c,k]
__global__ void output_kernel(const float* __restrict__ pooled,
                              const float* __restrict__ W_out,
                              const float* __restrict__ b_out,
                              float* __restrict__ out, int N) {
  int c = threadIdx.x;
  if (c < N_CLASS) {
    float invN = 1.0f / (float)N;
    float s = b_out[c];
    for (int k = 0; k < N_CENT; ++k)
      s += (pooled[k] * invN) * W_out[c * N_CENT + k];
    out[c] = s;
  }
}

// ---------------------------------------------------------------------------
extern "C" void kernel_launch(void* const* d_in, const int* in_sizes, int n_in,
                              void* d_out, int out_size, void* d_ws, size_t ws_size,
                              hipStream_t stream) {
  const float*     x       = (const float*)d_in[0];
  const long long* edge    = (const long long*)d_in[1];   // int64 [2, E]
  const float*     W_embed = (const float*)d_in[2];       // [64, 128]
  const float*     W_gcn   = (const float*)d_in[3];       // [L, 64, 64]
  const float*     b_gcn   = (const float*)d_in[4];       // [L, 64]
  const float*     cent    = (const float*)d_in[5];       // [100, 64]
  const float*     W_out   = (const float*)d_in[6];       // [10, 100]
  const float*     b_out   = (const float*)d_in[7];       // [10]
  float*           out     = (float*)d_out;

  const int IN_FEAT = 128;
  const int N = in_sizes[0] / IN_FEAT;                    // 100000
  const long long E = in_sizes[1] / 2;                    // 1.6M
  const int NL = in_sizes[3] / (EMBED_D * EMBED_D);       // 3

  // workspace carve-out
  char* ws = (char*)d_ws;
  size_t off = 0;
  auto carve = [&](size_t bytes) {
    size_t p = off;
    off += (bytes + 255) & ~(size_t)255;
    return p;
  };
  float* dinv   = (float*)(ws + carve((size_t)N * sizeof(float)));
  float* hbuf   = (float*)(ws + carve((size_t)N * EMBED_D * sizeof(float)));
  float* hwbuf  = (float*)(ws + carve((size_t)N * EMBED_D * sizeof(float)));
  float* accbuf = (float*)(ws + carve((size_t)N * EMBED_D * sizeof(float)));
  float* pooled = (float*)(ws + carve(128 * sizeof(float)));

  const long long* src = edge;
  const long long* dst = edge + E;

  // ---- degree -> dinv -------------------------------------------------------
  init_deg_kernel<<<(N + 255) / 256, 256, 0, stream>>>(dinv, N);
  accum_deg_kernel<<<(int)((E + 255) / 256), 256, 0, stream>>>(dst, dinv, E);
  finish_dinv_kernel<<<(N + 255) / 256, 256, 0, stream>>>(dinv, N);

  // ---- embed: h = x @ W_embed^T  (fp32 WMMA, K=128) -------------------------
  const int mtiles = (N + 15) / 16;
  const int gblocks = (mtiles + 7) / 8;
  gemm_nt64_wmma_kernel<128><<<gblocks, 256, 0, stream>>>(x, W_embed, hbuf, N);

  // ---- GCN layers -----------------------------------------------------------
  const long long scatter_threads = (E + (long long)N) << 6;
  const int scatter_blocks = (int)((scatter_threads + 255) / 256);
  const long long nd = (long long)N * EMBED_D;

  for (int l = 0; l < NL; ++l) {
    gemm_nt64_wmma_kernel<64><<<gblocks, 256, 0, stream>>>(
        hbuf, W_gcn + (size_t)l * EMBED_D * EMBED_D, hwbuf, N);
    hipMemsetAsync(accbuf, 0, (size_t)nd * sizeof(float), stream);
    scatter_edges_kernel<<<scatter_blocks, 256, 0, stream>>>(
        src, dst, dinv, hwbuf, accbuf, E, (long long)N);
    bias_relu_kernel<<<(int)((nd + 255) / 256), 256, 0, stream>>>(
        accbuf, b_gcn + (size_t)l * EMBED_D, hbuf, nd);
  }

  // ---- centroid distances + mean pool + classifier --------------------------
  hipMemsetAsync(pooled, 0, 128 * sizeof(float), stream);
  centroid_pool_kernel<<<(N + 255) / 256, 256, 0, stream>>>(hbuf, cent, pooled, N);
  output_kernel<<<1, 32, 0, stream>>>(pooled, W_out, b_out, out, N);
}